// CellNetwork_62036507623881
// MI455X (gfx1250) — compile-verified
//
#include <hip/hip_runtime.h>

// ---------------------------------------------------------------------------
// CellNetwork forward for MI455X (gfx1250, wave32).
// GEMMs: V_WMMA_F32_16X16X4_F32, 64-row blocks (4 M-tiles/wave) so each B
// fragment feeds 4 WMMAs -> 4x less W traffic from L2. ReLU is a template
// param; full-tile blocks take a uniform unguarded store path.
// Sparse aggregation: float4 gathers + coalesced global_atomic_add_f32.
// ---------------------------------------------------------------------------

#define FD 128                    // feature dim (fixed by reference)

typedef float v2f __attribute__((ext_vector_type(2)));
typedef float v8f __attribute__((ext_vector_type(8)));

// ---------------------------------------------------------------------------
// Y[tile64, :] = X[tile64, :] @ W  (K = N = 128), optional fused ReLU.
// Block = 256 threads = 8 waves; wave w owns columns [16w, 16w+16) for all
// four 16-row M-tiles. Ragged M: zero-fill staging + guarded tail store.
// ---------------------------------------------------------------------------
template <int RELU>
__global__ void __launch_bounds__(256)
gemm_d128_kernel(const float* __restrict__ X, const float* __restrict__ W,
                 float* __restrict__ Y, int M)
{
    __shared__ float sX[64 * FD];                 // 32 KB X tile

    const int t    = threadIdx.x;
    const int base = blockIdx.x * (64 * FD);      // element offset of tile
    const int lim  = M * FD;                      // < 2^31 for all uses

    // Cooperative stage of 64x128 X rows (2048 float4, 8 per thread).
    {
        const float4* __restrict__ g4 = (const float4*)(X + base);
        float4* s4 = (float4*)sX;
#pragma unroll
        for (int j = 0; j < 8; ++j) {
            const int i = t + j * 256;
            float4 v = make_float4(0.f, 0.f, 0.f, 0.f);
            if (base + i * 4 < lim) v = g4[i];
            s4[i] = v;
        }
    }
    __syncthreads();

    const int lane  = t & 31;
    const int wave  = t >> 5;
    const int m     = lane & 15;                  // A row / C column index
    const int koff  = (lane < 16) ? 0 : 2;        // K split across half-waves
    const int col   = wave * 16 + m;              // output column
    const int rhalf = (lane < 16) ? 0 : 8;        // C/D row half

    v8f acc[4] = {};
#pragma unroll 4
    for (int kc = 0; kc < 32; ++kc) {
        const int k0 = kc * 4 + koff;
        v2f b;
        b.x = W[k0 * FD + col];
        b.y = W[(k0 + 1) * FD + col];
#pragma unroll
        for (int r = 0; r < 4; ++r) {
            v2f a;
            a.x = sX[(r * 16 + m) * FD + k0];
            a.y = sX[(r * 16 + m) * FD + k0 + 1];
            // 8 args: (neg_a, A, neg_b, B, c_mod, C, reuse_a, reuse_b)
            acc[r] = __builtin_amdgcn_wmma_f32_16x16x4_f32(
                         false, a, false, b, (short)0, acc[r], false, false);
        }
    }

    const int rowbase = blockIdx.x * 64;
    if (rowbase + 64 <= M) {
        // Uniform fast path: full 64-row tile, unguarded stores.
#pragma unroll
        for (int r = 0; r < 4; ++r) {
#pragma unroll
            for (int v = 0; v < 8; ++v) {
                const int row = rowbase + r * 16 + v + rhalf;  // C/D layout
                float val = acc[r][v];
                if (RELU) val = fmaxf(val, 0.0f);
                Y[row * FD + col] = val;
            }
        }
    } else {
        // Tail block (at most one per GEMM): per-row guard, 32-bit compares.
#pragma unroll
        for (int r = 0; r < 4; ++r) {
#pragma unroll
            for (int v = 0; v < 8; ++v) {
                const int row = rowbase + r * 16 + v + rhalf;
                if (row < M) {
                    float val = acc[r][v];
                    if (RELU) val = fmaxf(val, 0.0f);
                    Y[row * FD + col] = val;
                }
            }
        }
    }
}

// ---------------------------------------------------------------------------
// dst[i] = src[i]  (float4 vectorized)
// ---------------------------------------------------------------------------
__global__ void copy_f4_kernel(const float4* __restrict__ src,
                               float4* __restrict__ dst, int n4)
{
    int i = blockIdx.x * blockDim.x + threadIdx.x;
    if (i < n4) dst[i] = src[i];
}

// ---------------------------------------------------------------------------
// tmp[dst[e], d] += h[src[e], d]   (GNN neighbor aggregation)
// One thread = one float4 (4 features); lanes sweep the feature dim.
// ---------------------------------------------------------------------------
__global__ void scatter_edges_kernel(const float* __restrict__ h,
                                     const int* __restrict__ src,
                                     const int* __restrict__ dst,
                                     float* __restrict__ tmp, int E)
{
    int i = blockIdx.x * blockDim.x + threadIdx.x;
    if (i >= E * (FD / 4)) return;
    const int e  = i >> 5;
    const int d4 = (i & 31) << 2;
    const float4 v = *(const float4*)&h[(size_t)src[e] * FD + d4];
    float* o = &tmp[(size_t)dst[e] * FD + d4];
    atomicAdd(o + 0, v.x);
    atomicAdd(o + 1, v.y);
    atomicAdd(o + 2, v.z);
    atomicAdd(o + 3, v.w);
}

// ---------------------------------------------------------------------------
// acc[idx0[k], d] += val[k] * T[idx1[k], d]   (COO SpMM scatter)
// ---------------------------------------------------------------------------
__global__ void spmm_scatter_kernel(const int* __restrict__ idx,
                                    const float* __restrict__ val,
                                    const float* __restrict__ T,
                                    float* __restrict__ acc, int nnz)
{
    int i = blockIdx.x * blockDim.x + threadIdx.x;
    if (i >= nnz * (FD / 4)) return;
    const int k  = i >> 5;
    const int d4 = (i & 31) << 2;
    const float s = val[k];
    const float4 v = *(const float4*)&T[(size_t)idx[nnz + k] * FD + d4];
    float* o = &acc[(size_t)idx[k] * FD + d4];
    atomicAdd(o + 0, s * v.x);
    atomicAdd(o + 1, s * v.y);
    atomicAdd(o + 2, s * v.z);
    atomicAdd(o + 3, s * v.w);
}

// ---------------------------------------------------------------------------
// p[i] = max(p[i], 0)  (float4 vectorized)
// ---------------------------------------------------------------------------
__global__ void relu_kernel(float4* __restrict__ p, int n4)
{
    int i = blockIdx.x * blockDim.x + threadIdx.x;
    if (i >= n4) return;
    float4 v = p[i];
    v.x = fmaxf(v.x, 0.f); v.y = fmaxf(v.y, 0.f);
    v.z = fmaxf(v.z, 0.f); v.w = fmaxf(v.w, 0.f);
    p[i] = v;
}

// ---------------------------------------------------------------------------
// out[n, 0:128] = h[n, :],  out[n, 128:256] = 0   (output init, float4)
// ---------------------------------------------------------------------------
__global__ void write_out_kernel(const float* __restrict__ h,
                                 float* __restrict__ out, int N)
{
    int i = blockIdx.x * blockDim.x + threadIdx.x;
    if (i >= N * (FD / 4)) return;
    const int n = i >> 5;
    const int d = i & 31;
    float4* o = (float4*)out;
    o[(size_t)n * 64 + d]      = ((const float4*)h)[(size_t)n * 32 + d];
    o[(size_t)n * 64 + 32 + d] = make_float4(0.f, 0.f, 0.f, 0.f);
}

// ---------------------------------------------------------------------------
// out[row[e], 128+d] += he[e,d];  out[col[e], 128+d] += he[e,d]
// ---------------------------------------------------------------------------
__global__ void final_scatter_kernel(const float* __restrict__ he,
                                     const int* __restrict__ row,
                                     const int* __restrict__ col,
                                     float* __restrict__ out, int E)
{
    int i = blockIdx.x * blockDim.x + threadIdx.x;
    if (i >= E * (FD / 4)) return;
    const int e  = i >> 5;
    const int d4 = (i & 31) << 2;
    const float4 v = *(const float4*)&he[(size_t)e * FD + d4];
    float* a = &out[(size_t)row[e] * (2 * FD) + FD + d4];
    float* b = &out[(size_t)col[e] * (2 * FD) + FD + d4];
    atomicAdd(a + 0, v.x); atomicAdd(a + 1, v.y);
    atomicAdd(a + 2, v.z); atomicAdd(a + 3, v.w);
    atomicAdd(b + 0, v.x); atomicAdd(b + 1, v.y);
    atomicAdd(b + 2, v.z); atomicAdd(b + 3, v.w);
}

// ---------------------------------------------------------------------------
// Host-side pipeline
// ---------------------------------------------------------------------------
extern "C" void kernel_launch(void* const* d_in, const int* in_sizes, int n_in,
                              void* d_out, int out_size, void* d_ws, size_t ws_size,
                              hipStream_t stream)
{
    const float* x       = (const float*)d_in[0];   // [N,128]
    const float* xe      = (const float*)d_in[1];   // [E,128]
    const float* gnn_W   = (const float*)d_in[2];   // [L,128,128]
    const float* cwnn_W0 = (const float*)d_in[3];
    const float* cwnn_W1 = (const float*)d_in[4];
    const float* cwnn_W2 = (const float*)d_in[5];
    const float* Ldo_val = (const float*)d_in[6];   // [NNZ]
    const float* Lup_val = (const float*)d_in[7];
    const int*   edges   = (const int*)d_in[8];     // [2,E]
    const int*   row     = (const int*)d_in[9];     // [E]
    const int*   col     = (const int*)d_in[10];
    const int*   Ldo_idx = (const int*)d_in[11];    // [2,NNZ]
    const int*   Lup_idx = (const int*)d_in[12];
    float* out = (float*)d_out;

    const int N   = in_sizes[0] / FD;
    const int E   = in_sizes[1] / FD;
    const int NNZ = in_sizes[6];
    const int L   = in_sizes[2] / (FD * FD);
    const int DD  = FD * FD;

    // Workspace carve-out (floats): h | tmp | heA | heB | T
    float* h_buf = (float*)d_ws;                    // N*FD
    float* tmp   = h_buf + (size_t)N * FD;          // N*FD
    float* heA   = tmp   + (size_t)N * FD;          // E*FD
    float* heB   = heA   + (size_t)E * FD;          // E*FD
    float* Tbuf  = heB   + (size_t)E * FD;          // E*FD

    const int TB = 256;
    const int nd4    = (N * FD) / 4;
    const int gnd4   = (nd4 + TB - 1) / TB;
    const int gE4    = (E * (FD / 4) + TB - 1) / TB;       // edge-wise f4 work
    const int gN4    = (N * (FD / 4) + TB - 1) / TB;
    const int gNNZ4  = (NNZ * (FD / 4) + TB - 1) / TB;
    const int gGemmN = (N + 63) / 64;
    const int gGemmE = (E + 63) / 64;
    const int gReluE = ((E * FD) / 4 + TB - 1) / TB;

    // ---- GNN over nodes: h = relu((h + scatter(h[src]->dst)) @ W_i) ----
    const float* cur = x;
    for (int i = 0; i < L; ++i) {
        copy_f4_kernel<<<gnd4, TB, 0, stream>>>((const float4*)cur,
                                                (float4*)tmp, nd4);
        scatter_edges_kernel<<<gE4, TB, 0, stream>>>(cur, edges, edges + E,
                                                     tmp, E);
        gemm_d128_kernel<1><<<gGemmN, TB, 0, stream>>>(tmp,
                                                       gnn_W + (size_t)i * DD,
                                                       h_buf, N);
        cur = h_buf;
    }

    // ---- CWNN over cells: he = relu(he@W0 + Ldo@(he@W1) + Lup@(he@W2)) ----
    const float* ce = xe;
    float* acc = heA;
    for (int i = 0; i < L; ++i) {
        acc = (i & 1) ? heB : heA;
        gemm_d128_kernel<0><<<gGemmE, TB, 0, stream>>>(ce,
                                                       cwnn_W0 + (size_t)i * DD,
                                                       acc, E);
        gemm_d128_kernel<0><<<gGemmE, TB, 0, stream>>>(ce,
                                                       cwnn_W1 + (size_t)i * DD,
                                                       Tbuf, E);
        spmm_scatter_kernel<<<gNNZ4, TB, 0, stream>>>(Ldo_idx, Ldo_val, Tbuf,
                                                      acc, NNZ);
        gemm_d128_kernel<0><<<gGemmE, TB, 0, stream>>>(ce,
                                                       cwnn_W2 + (size_t)i * DD,
                                                       Tbuf, E);
        spmm_scatter_kernel<<<gNNZ4, TB, 0, stream>>>(Lup_idx, Lup_val, Tbuf,
                                                      acc, NNZ);
        relu_kernel<<<gReluE, TB, 0, stream>>>((float4*)acc, (E * FD) / 4);
        ce = acc;
    }

    // ---- Assemble output: [h | segsum(he,row)+segsum(he,col)] ----
    write_out_kernel<<<gN4, TB, 0, stream>>>(cur, out, N);
    final_scatter_kernel<<<gE4, TB, 0, stream>>>(ce, row, col, out, E);
}